// G3_SF_7919919693927
// MI455X (gfx1250) — compile-verified
//
#include <hip/hip_runtime.h>
#include <stdint.h>

#define RC2_F 36.0f
#define PI_OVER_RC 0.52359877559829887f   // pi / 6

typedef __attribute__((ext_vector_type(2))) float        v2f;
typedef __attribute__((ext_vector_type(8))) float        v8f;
typedef __attribute__((ext_vector_type(4))) unsigned int v4ui;
typedef __attribute__((ext_vector_type(8))) int          v8i;
typedef __attribute__((ext_vector_type(4))) int          v4i;

// ---------------- CDNA5 async global->LDS path (ASYNCcnt-tracked) -----------
__device__ __forceinline__ void async_g2l_b64(uint32_t lds_byte_off, const void* gaddr) {
    asm volatile("global_load_async_to_lds_b64 %0, %1, off"
                 :: "v"(lds_byte_off), "v"(gaddr)
                 : "memory");
}
__device__ __forceinline__ void wait_async_le1() {
    asm volatile("s_wait_asynccnt 0x1" ::: "memory");
}
__device__ __forceinline__ void wait_async_le0() {
    asm volatile("s_wait_asynccnt 0x0" ::: "memory");
}

// ---------------- WMMA batch dot products -----------------------------------
// D = A(16x4) x B(4x16): A rows = dij of 16 triples, B cols = dik of 16 triples,
// K = {x,y,z,0}.  diag(D)[t] = dij_t . dik_t.
__device__ __forceinline__ v8f wmma_dot_group(float ax, float ay, float az,
                                              float bx, float by, float bz,
                                              int lane, int g) {
    int  s  = (g << 4) | (lane & 15);   // source lane holding triple (g*16 + lane%16)
    bool lo = lane < 16;
    // A 16x4 f32 layout: lanes0-15: {K0,K1}; lanes16-31: {K2,K3}
    float axs = __shfl(ax, s, 32), ays = __shfl(ay, s, 32), azs = __shfl(az, s, 32);
    v2f A; A.x = lo ? axs : azs;  A.y = lo ? ays : 0.0f;
    // B 4x16 f32 layout mirrors A: vgpr0 = rows K0(l0-15)/K2(l16-31), vgpr1 = K1/K3
    float bxs = __shfl(bx, s, 32), bys = __shfl(by, s, 32), bzs = __shfl(bz, s, 32);
    v2f B; B.x = lo ? bxs : bzs;  B.y = lo ? bys : 0.0f;
    v8f C = {};
    return __builtin_amdgcn_wmma_f32_16x16x4_f32(false, A, false, B, (short)0, C,
                                                 false, false);
}
// diag owner lanes: t<8 -> lane t (vgpr t); t>=8 -> lane t+16 (vgpr t-8).
// every lane packs d[lane&7]; consumer shuffles from the owner lane.
__device__ __forceinline__ float pack_diag(v8f d, int lane) {
    int r = lane & 7;
    float a01 = (r & 1) ? d[1] : d[0];
    float a23 = (r & 1) ? d[3] : d[2];
    float a45 = (r & 1) ? d[5] : d[4];
    float a67 = (r & 1) ? d[7] : d[6];
    float b03 = (r & 2) ? a23 : a01;
    float b47 = (r & 2) ? a67 : a45;
    return (r & 4) ? b47 : b03;
}

// ---------------- init: fp = 0, pair_i = -1 ---------------------------------
__global__ void sf_init_kernel(float* __restrict__ fp, int fp_n,
                               int* __restrict__ pairi, int n_pair) {
    int t = blockIdx.x * blockDim.x + threadIdx.x;
    int stride = gridDim.x * blockDim.x;
    for (int k = t; k < fp_n; k += stride) fp[k] = 0.0f;
    for (int k = t; k < n_pair; k += stride) pairi[k] = -1;
}

// ---------------- main triple kernel ----------------------------------------
__global__ void __launch_bounds__(256)
sf_triple_kernel(const int*   __restrict__ ind2,   // [n_pair][2]
                 const uint2* __restrict__ ind3,   // [n_tri][2]
                 const float* __restrict__ dist,   // [n_pair]
                 const float* __restrict__ diff,   // [n_pair][3]
                 const float* __restrict__ fc,     // [n_pair]
                 float* __restrict__ fp,           // [n_atom][16]
                 int*   __restrict__ pairi,        // [n_pair]
                 int n_tri)
{
    __shared__ uint2 lbuf[2][256];
    const int tid  = threadIdx.x;
    const int lane = tid & 31;
    const int ntiles = (n_tri + 255) >> 8;

    uint32_t lds_off0 = (uint32_t)(uintptr_t)(&lbuf[0][tid]);
    uint32_t lds_off1 = (uint32_t)(uintptr_t)(&lbuf[1][tid]);

    int tb = blockIdx.x;
    if (tb >= ntiles) return;

    {   // prologue: stage 0 async load (clamped => always valid indices in LDS)
        size_t e = ((size_t)tb << 8) + (size_t)tid;
        if (e >= (size_t)n_tri) e = (size_t)n_tri - 1;
        async_g2l_b64(lds_off0, (const void*)(ind3 + e));
    }

    int stage = 0;
    for (;;) {
        int tnext = tb + gridDim.x;
        bool more = (tnext < ntiles);
        if (more) {
            size_t e = ((size_t)tnext << 8) + (size_t)tid;
            if (e >= (size_t)n_tri) e = (size_t)n_tri - 1;
            async_g2l_b64(stage ? lds_off0 : lds_off1, (const void*)(ind3 + e));
            wait_async_le1();
        } else {
            wait_async_le0();
        }

        // ---- converged region: gathers + WMMA dots (EXEC all-1s for WMMA) ----
        uint2 pr = lbuf[stage][tid];
        int ij = (int)pr.x;
        int ik = (int)pr.y;

        float ax = diff[3*(size_t)ij+0], ay = diff[3*(size_t)ij+1], az = diff[3*(size_t)ij+2];
        float bx = diff[3*(size_t)ik+0], by = diff[3*(size_t)ik+1], bz = diff[3*(size_t)ik+2];
        float rij = dist[ij], rik = dist[ik];

        v8f D0 = wmma_dot_group(ax, ay, az, bx, by, bz, lane, 0);  // triples 0..15
        v8f D1 = wmma_dot_group(ax, ay, az, bx, by, bz, lane, 1);  // triples 16..31
        float pk0 = pack_diag(D0, lane);
        float pk1 = pack_diag(D1, lane);
        int   tt   = lane & 15;
        int   srcl = (tt < 8) ? tt : tt + 16;
        float dg0 = __shfl(pk0, srcl, 32);
        float dg1 = __shfl(pk1, srcl, 32);
        float dot = (lane < 16) ? dg0 : dg1;                 // dij . dik

        float rij2 = rij * rij, rik2 = rik * rik;
        float d2 = fmaxf(rij2 + rik2 - 2.0f * dot, 0.0f);    // |dik - dij|^2

        int t = (tb << 8) + tid;
        if (t < n_tri && d2 < RC2_F) {
            float rjk = __builtin_amdgcn_sqrtf(d2);          // raw v_sqrt_f32
            float fij = fc[ij], fik = fc[ik];
            float fjk = 0.5f * (__cosf(PI_OVER_RC * rjk) + 1.0f);
            float cosv = dot * __builtin_amdgcn_rcpf(rij * rik); // raw v_rcp_f32
            float S = rij2 + rik2 + d2;
            float C = fij * fik * fjk;

            float bm = fmaxf(1.0f - cosv, 0.0f);             // lambda = -1 (even s)
            float bp = fmaxf(1.0f + cosv, 0.0f);             // lambda = +1 (odd s)
            float pm[4], pp[4];                              // zeta = 1,2,4,8
            pm[0] = bm; pm[1] = bm*bm; pm[2] = pm[1]*pm[1]; pm[3] = pm[2]*pm[2];
            pp[0] = bp; pp[1] = bp*bp; pp[2] = pp[1]*pp[1]; pp[3] = pp[2]*pp[2];
            const float scl[4] = {1.0f, 0.5f, 0.125f, 0.0078125f}; // 2^(1-zeta)
            const float eta[16] = {0.01f,0.014f,0.02f,0.028f,0.04f,0.056f,0.08f,0.11f,
                                   0.16f,0.22f,0.32f,0.45f,0.63f,0.72f,0.9f,1.0f};

            int i = ind2[2*(size_t)ij];                      // i_rind
            float* row = fp + (size_t)i * 16;
            #pragma unroll
            for (int s = 0; s < 16; ++s) {
                int z = (s & 7) >> 1;
                float g  = scl[z] * ((s & 1) ? pp[z] : pm[z]);
                float sf = C * g * __expf(-eta[s] * S);
                unsafeAtomicAdd(row + s, sf);                // global_atomic_add_f32
            }
            atomicMax(&pairi[ij], i);                        // global_atomic_max_i32
            atomicMax(&pairi[ik], i);
        }

        if (!more) break;
        tb = tnext;
        stage ^= 1;
    }
}

// ---------------- jacob_ind writer (TDM-staged stream) ----------------------
#define JT 4096

__global__ void __launch_bounds__(256)
sf_jacob_kernel(const int* __restrict__ pairi, float* __restrict__ outj, int n_pair) {
    __shared__ int jbuf[JT];
    const int tile0 = blockIdx.x * JT;

#if __has_builtin(__builtin_amdgcn_tensor_load_to_lds)
    if (threadIdx.x < 32) {   // wave 0 drives the TDM (EXEC ignored by TDM)
        uint32_t lds = (uint32_t)(uintptr_t)(&jbuf[0]);
        uint64_t ga  = (uint64_t)(uintptr_t)(pairi + tile0);
        v4ui g0;
        g0[0] = 1u;                                        // count=1, user mode
        g0[1] = lds;                                       // lds_addr
        g0[2] = (uint32_t)ga;                              // global_addr[31:0]
        g0[3] = (uint32_t)((ga >> 32) & 0x01FFFFFFu) | 0x80000000u; // ga[56:32] | type=2
        v8i g1;
        g1[0] = (int)(2u << 16);                           // data_size=4B, no mcast/pad
        g1[1] = (int)(((uint32_t)n_pair & 0xFFFFu) << 16); // tensor_dim0[15:0] @bits63:48
        g1[2] = (int)((((uint32_t)n_pair >> 16) & 0xFFFFu) | (1u << 16)); // td0 hi | td1=1
        g1[3] = (int)((uint32_t)JT << 16);                 // tile_dim0 @bits127:112
        g1[4] = 1;                                         // tile_dim1=1, tile_dim2=0
        g1[5] = (int)(uint32_t)n_pair;                     // tensor_dim0_stride[31:0]
        g1[6] = 0;
        g1[7] = 0;
        v4i g2 = {0, 0, 0, 0};
        v4i g3 = {0, 0, 0, 0};
#if __clang_major__ >= 23
        v8i gx = {0, 0, 0, 0, 0, 0, 0, 0};
        __builtin_amdgcn_tensor_load_to_lds(g0, g1, g2, g3, gx, 0);
#else
        __builtin_amdgcn_tensor_load_to_lds(g0, g1, g2, g3, 0);
#endif
        __builtin_amdgcn_s_wait_tensorcnt(0);
    }
    __syncthreads();
    for (int k = threadIdx.x; k < JT; k += 256) {
        int p = tile0 + k;
        if (p < n_pair) {
            outj[2*(size_t)p + 0] = (float)p;
            outj[2*(size_t)p + 1] = (float)jbuf[k];
        }
    }
#else
    (void)jbuf;
    for (int k = threadIdx.x; k < JT; k += 256) {
        int p = tile0 + k;
        if (p < n_pair) {
            outj[2*(size_t)p + 0] = (float)p;
            outj[2*(size_t)p + 1] = (float)pairi[p];
        }
    }
#endif
}

// ---------------- launcher ---------------------------------------------------
extern "C" void kernel_launch(void* const* d_in, const int* in_sizes, int n_in,
                              void* d_out, int out_size, void* d_ws, size_t ws_size,
                              hipStream_t stream) {
    const int*   ind2 = (const int*)d_in[0];
    const uint2* ind3 = (const uint2*)d_in[1];
    const float* dist = (const float*)d_in[2];
    const float* diff = (const float*)d_in[3];
    /* d_in[4] = elems : unused by the reference computation */
    const float* fc   = (const float*)d_in[5];

    const int n_pair = in_sizes[2];        // dist
    const int n_tri  = in_sizes[1] / 2;    // ind_3
    const int n_atom = in_sizes[4];        // elems

    float* fp   = (float*)d_out;                       // [n_atom][16]
    float* outj = (float*)d_out + (size_t)n_atom * 16; // [n_pair][2]
    int*   pairi = (int*)d_ws;                         // [n_pair] scratch

    sf_init_kernel<<<1024, 256, 0, stream>>>(fp, n_atom * 16, pairi, n_pair);

    int ntiles = (n_tri + 255) / 256;
    int blocks = ntiles < 4096 ? ntiles : 4096;
    sf_triple_kernel<<<blocks, 256, 0, stream>>>(ind2, ind3, dist, diff, fc,
                                                 fp, pairi, n_tri);

    sf_jacob_kernel<<<(n_pair + JT - 1) / JT, 256, 0, stream>>>(pairi, outj, n_pair);
}